// GaussianModel_59493886984835
// MI455X (gfx1250) — compile-verified
//
#include <hip/hip_runtime.h>

// ---------------------------------------------------------------------------
// cov3d (Gaussian splatting 3D covariance) for MI455X / gfx1250.
// Memory-bound streaming kernel: 64 B/element, roofline ~11us @ 23.3 TB/s.
// Uses CDNA5 async global->LDS data mover (ASYNCcnt) for aligned b128 input
// staging, LDS for access-shape conversion, and non-temporal b128 stores.
// ---------------------------------------------------------------------------

#define TPB 256

typedef float v4f __attribute__((ext_vector_type(4)));
typedef int   v4i __attribute__((ext_vector_type(4)));
typedef __attribute__((address_space(1))) v4i* as1_v4i;   // global int4*
typedef __attribute__((address_space(3))) v4i* as3_v4i;   // LDS int4* (32-bit)

// ---- CDNA5 async global->LDS b128 copy (tracked by ASYNCcnt) --------------
// Signature (from hipcc diagnostic): (int4 AS1*, int4 AS3*, imm offset, imm cpol)
#if __has_builtin(__builtin_amdgcn_global_load_async_to_lds_b128)
#define ASYNC_G2L_B128(gp, lp)                                                \
  __builtin_amdgcn_global_load_async_to_lds_b128(                             \
      (as1_v4i)(unsigned long long)(gp),                                      \
      (as3_v4i)(unsigned)(unsigned long long)(lp), 0, 0)
#else
// Fallback: ISA mnemonic per CDNA5 ISA (opcode 98, GV mode).
#define ASYNC_G2L_B128(gp, lp)                                                \
  asm volatile("global_load_async_to_lds_b128 %0, %1, off"                    \
               :: "v"((unsigned)(unsigned long long)(lp)),                    \
                  "v"((const void*)(gp))                                      \
               : "memory")
#endif

#if __has_builtin(__builtin_amdgcn_s_wait_asynccnt)
#define WAIT_ASYNC0() __builtin_amdgcn_s_wait_asynccnt(0)
#else
#define WAIT_ASYNC0() asm volatile("s_wait_asynccnt 0x0" ::: "memory")
#endif

// ---- per-element math (matches the JAX reference) -------------------------
__device__ __forceinline__ void cov3d_one(float r0, float r1, float r2, float r3,
                                          float s0, float s1, float s2,
                                          float c[9]) {
  float nn  = r0 * r0 + r1 * r1 + r2 * r2 + r3 * r3;
  float d   = fmaxf(sqrtf(nn), 1e-12f);
  float inv = 1.0f / d;
  float w = r0 * inv, x = r1 * inv, y = r2 * inv, z = r3 * inv;
  float e0 = __expf(s0), e1 = __expf(s1), e2 = __expf(s2);

  float xx = x * x, yy = y * y, zz = z * z;
  float xy = x * y, xz = x * z, yz = y * z;
  float wx = w * x, wy = w * y, wz = w * z;

  float M00 = (1.0f - 2.0f * (yy + zz)) * e0;
  float M01 = (2.0f * (xy - wz)) * e1;
  float M02 = (2.0f * (xz + wy)) * e2;
  float M10 = (2.0f * (xy + wz)) * e0;
  float M11 = (1.0f - 2.0f * (xx + zz)) * e1;
  float M12 = (2.0f * (yz - wx)) * e2;
  float M20 = (2.0f * (xz - wy)) * e0;
  float M21 = (2.0f * (yz + wx)) * e1;
  float M22 = (1.0f - 2.0f * (xx + yy)) * e2;

  c[0] = M00 * M00 + M01 * M01 + M02 * M02;   // S00
  c[1] = M00 * M10 + M01 * M11 + M02 * M12;   // S01
  c[2] = M00 * M20 + M01 * M21 + M02 * M22;   // S02
  c[3] = c[1];                                // S10
  c[4] = M10 * M10 + M11 * M11 + M12 * M12;   // S11
  c[5] = M10 * M20 + M11 * M21 + M12 * M22;   // S12
  c[6] = c[2];                                // S20
  c[7] = c[5];                                // S21
  c[8] = M20 * M20 + M21 * M21 + M22 * M22;   // S22
}

__global__ __launch_bounds__(TPB)
void cov3d_kernel(const float* __restrict__ rot,
                  const float* __restrict__ scl,
                  float* __restrict__ out,
                  int n) {
  __shared__ float sIn[TPB * 4 + TPB * 3];  // 1024 rot + 768 scale floats (7 KB)
  __shared__ float sOut[TPB * 9];           // 2304 floats (9 KB)

  const int tid  = threadIdx.x;
  const int base = blockIdx.x * TPB;

  if (base + TPB <= n) {
    // ---------------- full tile: async-stage inputs to LDS ----------------
    const float* grot = rot + (size_t)base * 4;  // 4096 B tile = 256 x b128
    const float* gscl = scl + (size_t)base * 3;  // 3072 B tile = 192 x b128
    float* lrot = sIn;
    float* lscl = sIn + TPB * 4;

    // Each lane moves one aligned 16-B chunk; 448 chunks total.
    ASYNC_G2L_B128(grot + tid * 4, lrot + tid * 4);
    if (tid < (TPB * 12) / 16) {  // 192 chunks of scale
      ASYNC_G2L_B128(gscl + tid * 4, lscl + tid * 4);
    }
    WAIT_ASYNC0();       // this wave's async copies landed in LDS
    __syncthreads();     // make all waves' tiles visible

    // Rotation: one aligned ds b128 per lane. Scale: stride-3-dword reads
    // (gcd(3,64)=1 -> bank-conflict free).
    v4f r = *(const v4f*)(lrot + tid * 4);
    float c[9];
    cov3d_one(r.x, r.y, r.z, r.w,
              lscl[tid * 3 + 0], lscl[tid * 3 + 1], lscl[tid * 3 + 2], c);

    // Stride-9-dword LDS writes (gcd(9,64)=1 -> conflict free).
#pragma unroll
    for (int k = 0; k < 9; ++k) sOut[tid * 9 + k] = c[k];
    __syncthreads();

    // ------------- coalesced, aligned, non-temporal b128 stores -----------
    // Block output tile = 9216 B (16-B aligned: 9216 = 576*16).
    float* gout = out + (size_t)base * 9;
#pragma unroll
    for (int j = tid; j < (TPB * 9) / 4; j += TPB) {
      v4f v = *(const v4f*)(sOut + j * 4);
      __builtin_nontemporal_store(v, (v4f*)(gout + j * 4));
    }
  } else {
    // ---------------- tail tile: direct guarded path ----------------------
    int idx = base + tid;
    if (idx < n) {
      const float* rp = rot + (size_t)idx * 4;
      const float* sp = scl + (size_t)idx * 3;
      float c[9];
      cov3d_one(rp[0], rp[1], rp[2], rp[3], sp[0], sp[1], sp[2], c);
      float* op = out + (size_t)idx * 9;
#pragma unroll
      for (int k = 0; k < 9; ++k) op[k] = c[k];
    }
  }
}

extern "C" void kernel_launch(void* const* d_in, const int* in_sizes, int n_in,
                              void* d_out, int out_size, void* d_ws, size_t ws_size,
                              hipStream_t stream) {
  (void)n_in; (void)out_size; (void)d_ws; (void)ws_size;
  const float* rot = (const float*)d_in[0];   // (N, 4) fp32
  const float* scl = (const float*)d_in[1];   // (N, 3) fp32
  float* out = (float*)d_out;                 // (N, 3, 3) fp32

  const int n = in_sizes[0] / 4;              // N
  const int blocks = (n + TPB - 1) / TPB;     // 15625 for N = 4e6
  cov3d_kernel<<<blocks, TPB, 0, stream>>>(rot, scl, out, n);
}